// GNNModel_30872224924177
// MI455X (gfx1250) — compile-verified
//
#include <hip/hip_runtime.h>

typedef __attribute__((ext_vector_type(2))) float v2f;
typedef __attribute__((ext_vector_type(4))) float v4f;
typedef __attribute__((ext_vector_type(8))) float v8f;

// ---------------------------------------------------------------------------
// Degree / normalization kernels
// ---------------------------------------------------------------------------
__global__ __launch_bounds__(256) void k_deg_init(float* __restrict__ deg, int n) {
    int i = blockIdx.x * blockDim.x + threadIdx.x;
    if (i < n) deg[i] = 1.0f;  // self-loop contributes 1 to every node's degree
}

__global__ __launch_bounds__(256) void k_deg_accum(const long long* __restrict__ dstIdx,
                                                   float* __restrict__ deg, int nE) {
    int e = blockIdx.x * blockDim.x + threadIdx.x;
    if (e < nE) atomicAdd(&deg[dstIdx[e]], 1.0f);
}

__global__ __launch_bounds__(256) void k_deg_rsqrt(float* __restrict__ deg, int n) {
    int i = blockIdx.x * blockDim.x + threadIdx.x;
    if (i < n) {
        float d = deg[i];
        deg[i] = (d > 0.0f) ? rsqrtf(d) : 0.0f;
    }
}

// ---------------------------------------------------------------------------
// GEMM via V_WMMA_F32_16X16X4_F32 (f32 A/B, f32 accumulate -> exact precision)
// Block = 8 waves; all waves share one 128x64 B panel staged in LDS.
// One wave computes a 16 x 64 output tile (4 accumulators); K fixed at 128.
//   A layout (16x4 f32): lane 0-15 -> M=lane, K={k0,k0+1}; lane 16-31 -> K={k0+2,k0+3}
//   B layout (4x16 f32): lane 0-15 -> N=lane, K={k0,k0+1}; lane 16-31 -> K={k0+2,k0+3}
//   C layout: VGPR v -> M = v + 8*(lane>=16), N = lane&15
// LDS B row stride = 80 floats: rows 2 apart differ by 32 banks -> the two
// 16-lane halves of each ds_load_b32 hit disjoint bank sets (conflict-free).
// ---------------------------------------------------------------------------
#define B_STRIDE 80

template <int NT, bool RELU_IN>
__global__ __launch_bounds__(256) void k_gemm_wmma(const float* __restrict__ A,
                                                   const float* __restrict__ W,
                                                   float* __restrict__ C,
                                                   int mTiles, int Nout) {
    __shared__ float Blds[128 * B_STRIDE];   // 40 KB of the WGP's 320 KB LDS

    const int nBase = blockIdx.y * (NT * 16);

    // --- cooperative fill of the shared B panel (128 x 64 floats) -----------
    {
        const int t = threadIdx.x;
#pragma unroll
        for (int i = 0; i < 8; ++i) {
            int idx = t + i * 256;        // 0..2047 float4 slots
            int row = idx >> 4;           // K row: 0..127
            int c4  = (idx & 15) << 2;    // col: 0,4,...,60
            v4f v = *(const v4f*)(W + (size_t)row * Nout + nBase + c4);
            *(v4f*)(&Blds[row * B_STRIDE + c4]) = v;
        }
    }
    __syncthreads();

    const int lane  = threadIdx.x & 31;
    const int wave  = threadIdx.x >> 5;
    const int mTile = blockIdx.x * 8 + wave;
    if (mTile >= mTiles) return;               // uniform per wave -> EXEC all-1s

    const int col   = lane & 15;               // M index (A) / N index (B,C)
    const int kHalf = lane >> 4;               // 0: K pair {0,1}; 1: K pair {2,3}

    const float* aRow = A + (size_t)(mTile * 16 + col) * 128 + 2 * kHalf;

    v8f zero = {};
    v8f acc[NT];
#pragma unroll
    for (int j = 0; j < NT; ++j) acc[j] = zero;

    // --- K loop, software-pipelined A fetch ---------------------------------
    v2f aCur = *(const v2f*)(aRow);
#pragma unroll 4
    for (int k0 = 0; k0 < 128; k0 += 4) {
        const int kn = (k0 + 4) & 127;         // wrap: last prefetch is redundant
        v2f aNext = *(const v2f*)(aRow + kn);

        v2f a = aCur;
        if (RELU_IN) { a.x = fmaxf(a.x, 0.0f); a.y = fmaxf(a.y, 0.0f); }

        const float* bRow = &Blds[(k0 + 2 * kHalf) * B_STRIDE + col];
#pragma unroll
        for (int j = 0; j < NT; ++j) {
            v2f b; b.x = bRow[j * 16]; b.y = bRow[j * 16 + B_STRIDE];
            acc[j] = __builtin_amdgcn_wmma_f32_16x16x4_f32(
                false, a, false, b, (short)0, acc[j], false, false);
        }
        aCur = aNext;
    }

    // --- store C tile --------------------------------------------------------
    const int cRow = mTile * 16 + 8 * kHalf;
#pragma unroll
    for (int j = 0; j < NT; ++j) {
        float* cp = C + (size_t)cRow * Nout + nBase + j * 16 + col;
#pragma unroll
        for (int v = 0; v < 8; ++v)
            cp[(size_t)v * Nout] = acc[j][v];
    }
}

// ---------------------------------------------------------------------------
// AGG[i][c] = bias[c] + H[i][c] * dis[i]^2   (self-loop folded, non-atomic)
// ---------------------------------------------------------------------------
template <int CCH>
__global__ __launch_bounds__(256) void k_self_bias(const float* __restrict__ H,
                                                   const float* __restrict__ bias,
                                                   const float* __restrict__ dis,
                                                   float* __restrict__ AGG, int n) {
    int t = blockIdx.x * blockDim.x + threadIdx.x;
    if (t >= n * CCH) return;
    int node = t / CCH;
    int c    = t - node * CCH;
    float dv = dis[node];
    AGG[t] = bias[c] + H[t] * dv * dv;
}

// ---------------------------------------------------------------------------
// Edge scatter: one 32-lane wave per edge; each lane owns VEC contiguous
// channels (vectorized load, float atomic adds into L2).
// ---------------------------------------------------------------------------
template <int CCH, int VEC>
__global__ __launch_bounds__(256) void k_scatter(const long long* __restrict__ srcIdx,
                                                 const long long* __restrict__ dstIdx,
                                                 const float* __restrict__ dis,
                                                 const float* __restrict__ H,
                                                 float* __restrict__ AGG, int nE) {
    int t = blockIdx.x * blockDim.x + threadIdx.x;
    int e    = t >> 5;
    int lane = t & 31;
    if (e >= nE) return;
    long long s = srcIdx[e];
    long long d = dstIdx[e];
    float norm = dis[s] * dis[d];
    const float* hp = H + (size_t)s * CCH + lane * VEC;
    float* ap = AGG + (size_t)d * CCH + lane * VEC;
    float vals[VEC];
#pragma unroll
    for (int i = 0; i < VEC; ++i) vals[i] = hp[i];
#pragma unroll
    for (int i = 0; i < VEC; ++i) atomicAdd(ap + i, vals[i] * norm);
}

// ---------------------------------------------------------------------------
// Launch
// ---------------------------------------------------------------------------
extern "C" void kernel_launch(void* const* d_in, const int* in_sizes, int n_in,
                              void* d_out, int out_size, void* d_ws, size_t ws_size,
                              hipStream_t stream) {
    const float*     x  = (const float*)d_in[0];
    const long long* ei = (const long long*)d_in[1];   // int64 [2, E] row-major
    const float*     W1 = (const float*)d_in[2];
    const float*     b1 = (const float*)d_in[3];
    const float*     W2 = (const float*)d_in[4];
    const float*     b2 = (const float*)d_in[5];

    const int CIN = 128, CHID = 128, COUT = 64;
    const int n  = in_sizes[0] / CIN;     // 100000
    const int nE = in_sizes[1] / 2;       // 1600000
    const long long* srcIdx = ei;
    const long long* dstIdx = ei + nE;

    // Workspace layout: dis[n] | bufA (n*128 f32) | bufB (n*128 f32)
    char*  ws    = (char*)d_ws;
    size_t offA  = ((size_t)n * sizeof(float) + 4095) & ~(size_t)4095;
    size_t bufSz = ((size_t)n * CHID * sizeof(float) + 4095) & ~(size_t)4095;
    float* dis  = (float*)ws;
    float* bufA = (float*)(ws + offA);          // h (N x 128), then h2 (N x 64)
    float* bufB = (float*)(ws + offA + bufSz);  // agg1 (N x 128)
    float* out  = (float*)d_out;

    dim3 blk(256);

    // --- symmetric normalization: dis = rsqrt(in-degree + 1) -----------------
    k_deg_init <<<dim3((n  + 255) / 256), blk, 0, stream>>>(dis, n);
    k_deg_accum<<<dim3((nE + 255) / 256), blk, 0, stream>>>(dstIdx, dis, nE);
    k_deg_rsqrt<<<dim3((n  + 255) / 256), blk, 0, stream>>>(dis, n);

    const int mTiles = n / 16;  // 100000 / 16 = 6250, exact

    // --- layer 1: h = x @ W1 -------------------------------------------------
    k_gemm_wmma<4, false><<<dim3((mTiles + 7) / 8, CHID / 64), blk, 0, stream>>>(
        x, W1, bufA, mTiles, CHID);
    // agg1 = b1 + self-loop term
    k_self_bias<128><<<dim3((n * CHID + 255) / 256), blk, 0, stream>>>(
        bufA, b1, dis, bufB, n);
    // agg1[dst] += h[src] * norm
    k_scatter<128, 4><<<dim3((nE * 32 + 255) / 256), blk, 0, stream>>>(
        srcIdx, dstIdx, dis, bufA, bufB, nE);

    // --- layer 2: h2 = relu(agg1) @ W2 (ReLU fused into A-load) --------------
    k_gemm_wmma<4, true><<<dim3((mTiles + 7) / 8, COUT / 64), blk, 0, stream>>>(
        bufB, W2, bufA, mTiles, COUT);
    // out = b2 + self-loop term
    k_self_bias<64><<<dim3((n * COUT + 255) / 256), blk, 0, stream>>>(
        bufA, b2, dis, out, n);
    // out[dst] += h2[src] * norm
    k_scatter<64, 2><<<dim3((nE * 32 + 255) / 256), blk, 0, stream>>>(
        srcIdx, dstIdx, dis, bufA, out, nE);
}